// RelativeMultiHeadAttn_81973745811494
// MI455X (gfx1250) — compile-verified
//
#include <hip/hip_runtime.h>
#include <hip/hip_bf16.h>
#include <math.h>
#include <stdint.h>

// ---------------------------------------------------------------------------
// TENER RelativeMultiHeadAttn for MI455X (gfx1250).
// f32 WMMA (v_wmma_f32_16x16x4_f32) + async global->LDS double buffering
// (global_load_async_to_lds_b128 / s_wait_asynccnt).
// score[b,h,q,k] = (q+rr)·k + (q+rw)·pos[k-q+512]   (shift-trick identity)
// ---------------------------------------------------------------------------

#define BATCH 8
#define LEN   512
#define DMODEL 1024
#define NHEAD 16
#define HD    64

typedef __attribute__((ext_vector_type(2))) float v2f;
typedef __attribute__((ext_vector_type(8))) float v8f;

__device__ __forceinline__ v8f wmma_f32(v2f a, v2f b, v8f c) {
    return __builtin_amdgcn_wmma_f32_16x16x4_f32(false, a, false, b, (short)0, c,
                                                 false, false);
}

// async 16B global -> LDS copy (per active lane), tracked by ASYNCcnt
__device__ __forceinline__ void async_ld16(const float* g, float* l) {
    asm volatile("global_load_async_to_lds_b128 %0, %1, off"
                 :: "v"((uint32_t)(uintptr_t)l), "v"((uint64_t)(uintptr_t)g)
                 : "memory");
}
#define WAIT_ASYNC(n) asm volatile("s_wait_asynccnt %0" :: "n"(n) : "memory")

__device__ __forceinline__ v8f zero8() {
    return v8f{0.f, 0.f, 0.f, 0.f, 0.f, 0.f, 0.f, 0.f};
}

// ---------------------------------------------------------------------------
// Kernel 0: sinusoidal relative position table  pos[1024][64]
// ---------------------------------------------------------------------------
__global__ void pos_kernel(float* __restrict__ pos) {
    int j = blockIdx.x;          // 0..1023  (rel position j-512)
    int d = threadIdx.x;         // 0..63
    int i = d & 31;
    float invf = expf(-(logf(10000.0f) / 31.0f) * (float)i);
    float ang  = (float)(j - 512) * invf;
    pos[j * HD + d] = (d < 32) ? sinf(ang) : cosf(ang);
}

// ---------------------------------------------------------------------------
// Kernel 1: qv = x(4096x1024) @ W(1024x2048), f32 WMMA, async double-buffered.
// Tile 64(M) x 128(N), BK=64. 256 threads = 8 wave32 (4x2 grid).
// ---------------------------------------------------------------------------
#define A_TILE (64 * 68)     // stride 68 floats: 16B-aligned rows, clean banks
#define B_TILE (64 * 132)    // stride 132
#define GEMM_SMEM_FLOATS (2 * A_TILE + 2 * B_TILE)   // 25600 f = 100 KB

__device__ __forceinline__ void gemm_stage(const float* __restrict__ x,
                                           const float* __restrict__ w,
                                           float* smg, int buf,
                                           int m0, int n0, int k0, int tid) {
    float* As = smg + buf * A_TILE;
    float* Bs = smg + 2 * A_TILE + buf * B_TILE;
    #pragma unroll
    for (int i = 0; i < 4; ++i) {           // A: 64x64 floats
        int f = tid + i * 256;
        int row = f >> 4, c4 = (f & 15) * 4;
        async_ld16(&x[(size_t)(m0 + row) * DMODEL + k0 + c4], &As[row * 68 + c4]);
    }
    #pragma unroll
    for (int i = 0; i < 2; ++i) {           // B: 16x128 floats
        int f = tid + i * 256;
        int row = f >> 5, c4 = (f & 31) * 4;
        async_ld16(&w[(size_t)(k0 + row) * 2048 + n0 + c4], &Bs[row * 132 + c4]);
    }
}

__global__ void qv_gemm_kernel(const float* __restrict__ x,
                               const float* __restrict__ w,
                               float* __restrict__ q_ws,
                               float* __restrict__ v_ws) {
    extern __shared__ float smg[];

    const int tid  = threadIdx.x;
    const int lane = tid & 31;
    const int wv   = tid >> 5;
    const int hi   = lane >> 4;
    const int nl   = lane & 15;

    const int m0 = blockIdx.x * 64;
    const int n0 = blockIdx.y * 128;
    const int wm = (wv >> 1) * 16;
    const int wn = (wv & 1) * 64;

    v8f acc[4];
    #pragma unroll
    for (int j = 0; j < 4; ++j) acc[j] = zero8();

    gemm_stage(x, w, smg, 0, m0, n0, 0, tid);

    for (int it = 0; it < 16; ++it) {
        if (it < 15) {
            gemm_stage(x, w, smg, (it + 1) & 1, m0, n0, (it + 1) * 64, tid);
            WAIT_ASYNC(6);      // previous stage's 6 copies have landed
        } else {
            WAIT_ASYNC(0);
        }
        __syncthreads();

        const float* As = smg + (it & 1) * A_TILE;
        const float* Bs = smg + 2 * A_TILE + (it & 1) * B_TILE;
        #pragma unroll 4
        for (int kk = 0; kk < 64; kk += 4) {
            int kbi = kk + 2 * hi;
            int mm = wm + nl;
            v2f a; a.x = As[mm * 68 + kbi]; a.y = As[mm * 68 + kbi + 1];
            #pragma unroll
            for (int j = 0; j < 4; ++j) {
                int nn = wn + j * 16 + nl;
                v2f b; b.x = Bs[kbi * 132 + nn]; b.y = Bs[(kbi + 1) * 132 + nn];
                acc[j] = wmma_f32(a, b, acc[j]);
            }
        }
        __syncthreads();        // all reads done before this buffer is restaged
    }

    // scatter into (B,H,L,64)
    #pragma unroll
    for (int j = 0; j < 4; ++j) {
        #pragma unroll
        for (int r = 0; r < 8; ++r) {
            int gm = m0 + wm + r + 8 * hi;
            int gn = n0 + wn + j * 16 + nl;
            int bb = gm >> 9, ll = gm & 511;
            float val = acc[j][r];
            if (gn < 1024) {
                int hh = gn >> 6, dd = gn & 63;
                q_ws[(((size_t)bb * NHEAD + hh) * LEN + ll) * HD + dd] = val;
            } else {
                int gn2 = gn - 1024;
                int hh = gn2 >> 6, dd = gn2 & 63;
                v_ws[(((size_t)bb * NHEAD + hh) * LEN + ll) * HD + dd] = val;
            }
        }
    }
}

// ---------------------------------------------------------------------------
// Kernel 2: fused relative attention, 1 WG per (b,h,64-row q-tile).
// Double-buffered async K/P staging; V prefetched under the softmax.
// ---------------------------------------------------------------------------
#define ST64  68
#define ST128 132
#define SSTR  516

#define OFF_S    0                         // 64 x 516 score strip
#define OFF_QRR  (OFF_S   + 64 * SSTR)     // 64 x 68
#define OFF_QRW  (OFF_QRR + 64 * ST64)
#define OFF_K0   (OFF_QRW + 64 * ST64)     // K double buffer (V buffers in ph.2)
#define OFF_K1   (OFF_K0  + 64 * ST64)
#define OFF_P0   (OFF_K1  + 64 * ST64)     // 128 x 68 pos-window double buffer
#define OFF_P1   (OFF_P0  + 128 * ST64)
#define OFF_G    (OFF_P1  + 128 * ST64)    // 64 x 132 pos band
#define OFF_MASK (OFF_G   + 64 * ST128)    // 512
#define OFF_RED  (OFF_MASK + 512)          // 256
#define ATTN_SMEM_FLOATS (OFF_RED + 256)   // 77056 f = 308224 B (< 320 KB/WGP)

__device__ __forceinline__ void stage_kp(const float* __restrict__ x,
                                         const float* __restrict__ pos,
                                         float* sm, int kb,
                                         int b, int h, int q0, int tid) {
    const int k0 = kb * 64;
    float* Ks = sm + ((kb & 1) ? OFF_K1 : OFF_K0);
    float* Ps = sm + ((kb & 1) ? OFF_P1 : OFF_P0);
    const float* kbase = x + ((size_t)b * LEN + k0) * DMODEL + h * HD;
    #pragma unroll
    for (int i = 0; i < 4; ++i) {          // K: 64x64
        int f = tid + i * 256;
        int row = f >> 4, c4 = (f & 15) * 4;
        async_ld16(&kbase[(size_t)row * DMODEL + c4], &Ks[row * ST64 + c4]);
    }
    const int w0 = k0 - q0 + 449;          // window start, j in [1, 1024]
    #pragma unroll
    for (int i = 0; i < 8; ++i) {          // pos window: 128x64
        int f = tid + i * 256;
        int row = f >> 4, c4 = (f & 15) * 4;
        int j = w0 + row;
        if (j > 1023) j = 1023;            // c=127 of G is never gathered
        async_ld16(&pos[(size_t)j * HD + c4], &Ps[row * ST64 + c4]);
    }
}

__device__ __forceinline__ void stage_v(const float* __restrict__ vb,
                                        float* sm, int kb, int tid) {
    float* Vs = sm + ((kb & 1) ? OFF_K1 : OFF_K0);
    #pragma unroll
    for (int i = 0; i < 4; ++i) {          // V: 64x64
        int f = tid + i * 256;
        int row = f >> 4, c4 = (f & 15) * 4;
        async_ld16(&vb[(size_t)(kb * 64 + row) * HD + c4], &Vs[row * ST64 + c4]);
    }
}

__global__ void attn_kernel(const float* __restrict__ x,
                            const float* __restrict__ mask,
                            const float* __restrict__ q_ws,
                            const float* __restrict__ v_ws,
                            const float* __restrict__ pos,
                            const float* __restrict__ rr_bias,
                            const float* __restrict__ rw_bias,
                            float* __restrict__ out) {
    extern __shared__ float sm[];

    const int tid  = threadIdx.x;
    const int lane = tid & 31;
    const int wv   = tid >> 5;
    const int hi   = lane >> 4;
    const int nl   = lane & 15;

    const int qt = blockIdx.x & 7;
    const int h  = (blockIdx.x >> 3) & 15;
    const int b  = blockIdx.x >> 7;
    const int q0 = qt * 64;

    const int ms = (wv & 3) * 16;     // wave's q-row strip
    const int ns = (wv >> 2) * 32;    // wave's 32-col half (scores / output)
    const int ch = (wv >> 2) * 64;    // wave's G col half

    // kick off K/P for block 0; overlap with Q-tile + mask staging below
    stage_kp(x, pos, sm, 0, b, h, q0, tid);

    {   // Q tile with both biases (sync loads, overlap async above)
        const float* qb  = q_ws + (((size_t)b * NHEAD + h) * LEN + q0) * HD;
        const float* rrb = rr_bias + h * HD;
        const float* rwb = rw_bias + h * HD;
        #pragma unroll
        for (int i = 0; i < 4; ++i) {
            int f = tid + i * 256;
            int row = f >> 4, c4 = (f & 15) * 4;
            float4 qv4 = *(const float4*)&qb[row * HD + c4];
            float4 rr4 = *(const float4*)&rrb[c4];
            float4 rw4 = *(const float4*)&rwb[c4];
            float4 a = {qv4.x + rr4.x, qv4.y + rr4.y, qv4.z + rr4.z, qv4.w + rr4.w};
            float4 c = {qv4.x + rw4.x, qv4.y + rw4.y, qv4.z + rw4.z, qv4.w + rw4.w};
            *(float4*)&sm[OFF_QRR + row * ST64 + c4] = a;
            *(float4*)&sm[OFF_QRW + row * ST64 + c4] = c;
        }
        if (tid < 128)
            *(float4*)&sm[OFF_MASK + tid * 4] =
                ((const float4*)(mask + (size_t)b * LEN))[tid];
    }

    // =========== Phase 1: score strip S[64][512] ===========
    for (int kb = 0; kb < 8; ++kb) {
        const int k0 = kb * 64;
        if (kb < 7) {
            stage_kp(x, pos, sm, kb + 1, b, h, q0, tid);
            WAIT_ASYNC(12);     // block kb's 12 copies have landed
        } else {
            WAIT_ASYNC(0);
        }
        __syncthreads();

        const float* Ks = sm + ((kb & 1) ? OFF_K1 : OFF_K0);
        const float* Ps = sm + ((kb & 1) ? OFF_P1 : OFF_P0);

        // G[m][c] = (q+rw)[m] · pos[w0+c]   (64 x 128, K=64)
        {
            v8f g[4];
            #pragma unroll
            for (int j = 0; j < 4; ++j) g[j] = zero8();
            #pragma unroll 4
            for (int kk = 0; kk < 64; kk += 4) {
                int kbi = kk + 2 * hi;
                int mm = ms + nl;
                v2f a; a.x = sm[OFF_QRW + mm * ST64 + kbi];
                       a.y = sm[OFF_QRW + mm * ST64 + kbi + 1];
                #pragma unroll
                for (int j = 0; j < 4; ++j) {
                    int cc = ch + j * 16 + nl;
                    v2f bb; bb.x = Ps[cc * ST64 + kbi];
                            bb.y = Ps[cc * ST64 + kbi + 1];
                    g[j] = wmma_f32(a, bb, g[j]);
                }
            }
            #pragma unroll
            for (int j = 0; j < 4; ++j)
                #pragma unroll
                for (int r = 0; r < 8; ++r) {
                    int row = ms + r + 8 * hi;
                    int col = ch + j * 16 + nl;
                    sm[OFF_G + row * ST128 + col] = g[j][r];
                }
        }
        __syncthreads();

        // S tile: (q+rr)·k + band-gather(G), mask, ==0 -> -inf
        {
            v8f s[2];
            s[0] = zero8(); s[1] = zero8();
            #pragma unroll 4
            for (int kk = 0; kk < 64; kk += 4) {
                int kbi = kk + 2 * hi;
                int mm = ms + nl;
                v2f a; a.x = sm[OFF_QRR + mm * ST64 + kbi];
                       a.y = sm[OFF_QRR + mm * ST64 + kbi + 1];
                #pragma unroll
                for (int j = 0; j < 2; ++j) {
                    int nn = ns + j * 16 + nl;
                    v2f bb; bb.x = Ks[nn * ST64 + kbi];
                            bb.y = Ks[nn * ST64 + kbi + 1];
                    s[j] = wmma_f32(a, bb, s[j]);
                }
            }
            #pragma unroll
            for (int j = 0; j < 2; ++j)
                #pragma unroll
                for (int r = 0; r < 8; ++r) {
                    int m = ms + r + 8 * hi;
                    int n = ns + j * 16 + nl;
                    float val = s[j][r] + sm[OFF_G + m * ST128 + (n - m + 63)];
                    val *= sm[OFF_MASK + k0 + n];
                    if (val == 0.0f) val = -__builtin_huge_valf();
                    sm[OFF_S + m * SSTR + k0 + n] = val;
                }
        }
        __syncthreads();        // readers done before buffers restaged
    }

    // prefetch V block 0 underneath the softmax
    const float* vb = v_ws + (((size_t)b * NHEAD + h) * LEN) * HD;
    stage_v(vb, sm, 0, tid);

    // =========== softmax over 512 cols (4 threads per row) ===========
    {
        const int r = tid >> 2, seg = tid & 3;
        float* Srow = &sm[OFF_S + r * SSTR + seg * 128];
        float lmax = -__builtin_huge_valf();
        for (int c = 0; c < 128; ++c) lmax = fmaxf(lmax, Srow[c]);
        sm[OFF_RED + tid] = lmax;
        __syncthreads();
        float rmax = fmaxf(fmaxf(sm[OFF_RED + r * 4 + 0], sm[OFF_RED + r * 4 + 1]),
                           fmaxf(sm[OFF_RED + r * 4 + 2], sm[OFF_RED + r * 4 + 3]));
        __syncthreads();
        float lsum = 0.f;
        for (int c = 0; c < 128; ++c) {
            float e = expf(Srow[c] - rmax);
            Srow[c] = e;
            lsum += e;
        }
        sm[OFF_RED + tid] = lsum;
        __syncthreads();
        float inv = 1.0f / (sm[OFF_RED + r * 4 + 0] + sm[OFF_RED + r * 4 + 1] +
                            sm[OFF_RED + r * 4 + 2] + sm[OFF_RED + r * 4 + 3]);
        for (int c = 0; c < 128; ++c) Srow[c] *= inv;
    }

    // =========== Phase 2: O = P @ V  (64 x 64, K=512) ===========
    v8f o[2];
    o[0] = zero8(); o[1] = zero8();
    for (int kb = 0; kb < 8; ++kb) {
        const int k0 = kb * 64;
        if (kb < 7) {
            stage_v(vb, sm, kb + 1, tid);
            WAIT_ASYNC(4);
        } else {
            WAIT_ASYNC(0);
        }
        __syncthreads();        // (iter 0: also publishes softmax result)

        const float* Vs = sm + ((kb & 1) ? OFF_K1 : OFF_K0);
        #pragma unroll 4
        for (int kk = 0; kk < 64; kk += 4) {
            int kbi = kk + 2 * hi;
            int mm = ms + nl;
            v2f a; a.x = sm[OFF_S + mm * SSTR + k0 + kbi];
                   a.y = sm[OFF_S + mm * SSTR + k0 + kbi + 1];
            #pragma unroll
            for (int j = 0; j < 2; ++j) {
                int nn = ns + j * 16 + nl;
                v2f bb; bb.x = Vs[kbi * ST64 + nn];
                        bb.y = Vs[(kbi + 1) * ST64 + nn];
                o[j] = wmma_f32(a, bb, o[j]);
            }
        }
        __syncthreads();
    }

    // write out[b, q0+m, h*64+n]
    float* ob = out + ((size_t)b * LEN + q0) * DMODEL + h * HD;
    #pragma unroll
    for (int j = 0; j < 2; ++j)
        #pragma unroll
        for (int r = 0; r < 8; ++r) {
            int m = ms + r + 8 * hi;
            int n = ns + j * 16 + nl;
            ob[(size_t)m * DMODEL + n] = o[j][r];
        }
}

// ---------------------------------------------------------------------------
extern "C" void kernel_launch(void* const* d_in, const int* in_sizes, int n_in,
                              void* d_out, int out_size, void* d_ws, size_t ws_size,
                              hipStream_t stream) {
    const float* x    = (const float*)d_in[0];   // (8,512,1024)
    const float* mask = (const float*)d_in[1];   // (8,512)
    const float* W    = (const float*)d_in[2];   // (1024,2048)
    const float* rr   = (const float*)d_in[3];   // (16,64)
    const float* rw   = (const float*)d_in[4];   // (16,64)
    float* out = (float*)d_out;

    float* q_ws   = (float*)d_ws;
    float* v_ws   = q_ws + (size_t)BATCH * NHEAD * LEN * HD;
    float* pos_ws = v_ws + (size_t)BATCH * NHEAD * LEN * HD;

    constexpr size_t GEMM_SMEM_BYTES = (size_t)GEMM_SMEM_FLOATS * sizeof(float);
    constexpr size_t ATTN_SMEM_BYTES = (size_t)ATTN_SMEM_FLOATS * sizeof(float);
    (void)hipFuncSetAttribute((const void*)qv_gemm_kernel,
                              hipFuncAttributeMaxDynamicSharedMemorySize,
                              (int)GEMM_SMEM_BYTES);
    (void)hipFuncSetAttribute((const void*)attn_kernel,
                              hipFuncAttributeMaxDynamicSharedMemorySize,
                              (int)ATTN_SMEM_BYTES);

    pos_kernel<<<1024, 64, 0, stream>>>(pos_ws);
    qv_gemm_kernel<<<dim3(4096 / 64, 2048 / 128), 256, GEMM_SMEM_BYTES, stream>>>(
        x, W, q_ws, v_ws);
    attn_kernel<<<BATCH * NHEAD * (LEN / 64), 256, ATTN_SMEM_BYTES, stream>>>(
        x, mask, q_ws, v_ws, pos_ws, rr, rw, out);
}